// HybridMemory_83253646065769
// MI455X (gfx1250) — compile-verified
//
#include <hip/hip_runtime.h>
#include <hip/hip_bf16.h>

// Problem constants (match reference)
#define B_DIM 1024
#define D_DIM 512
#define M_DIM 65536
#define C_DIM 4096
#define TEMP_F 0.05f
#define EPS_F 1e-6f

typedef __attribute__((ext_vector_type(2))) float v2f;
typedef __attribute__((ext_vector_type(8))) float v8f;

// ---------------------------------------------------------------------------
// Kernel 0: zero the accumulators (Fc, nums) and the scalar output.
// ---------------------------------------------------------------------------
__global__ __launch_bounds__(256) void zero_ws(float* __restrict__ fc,
                                               float* __restrict__ nums,
                                               float* __restrict__ out) {
  int i = blockIdx.x * 256 + threadIdx.x;
  int total = C_DIM * D_DIM + C_DIM;
  if (i < C_DIM * D_DIM) {
    fc[i] = 0.0f;
  } else if (i < total) {
    nums[i - C_DIM * D_DIM] = 0.0f;
  }
  if (i == 0) out[0] = 0.0f;
}

// ---------------------------------------------------------------------------
// Kernel 1: L2-normalize each input row. One 256-thread block per row.
// ---------------------------------------------------------------------------
__global__ __launch_bounds__(256) void normalize_rows(const float* __restrict__ in,
                                                      float* __restrict__ out) {
  __shared__ float red[8];
  const int b = blockIdx.x;
  const float* row = in + (size_t)b * D_DIM;
  float s = 0.0f;
  for (int d = threadIdx.x; d < D_DIM; d += 256) {
    float v = row[d];
    s += v * v;
  }
  // wave32 reduction
  for (int off = 16; off > 0; off >>= 1) s += __shfl_down(s, off, 32);
  const int lane = threadIdx.x & 31;
  const int wave = threadIdx.x >> 5;
  if (lane == 0) red[wave] = s;
  __syncthreads();
  if (threadIdx.x == 0) {
    float t = 0.0f;
#pragma unroll
    for (int i = 0; i < 8; ++i) t += red[i];
    red[0] = 1.0f / sqrtf(t);
  }
  __syncthreads();
  const float inv = red[0];
  for (int d = threadIdx.x; d < D_DIM; d += 256) out[(size_t)b * D_DIM + d] = row[d] * inv;
}

// ---------------------------------------------------------------------------
// Kernel 2: segment-sum feature rows into per-cluster sums + counts.
// One block per memory row; streams 128 MB once, atomics land in L2 (Fc=8MB).
// ---------------------------------------------------------------------------
__global__ __launch_bounds__(256) void cluster_accum(const float* __restrict__ feats,
                                                     const int* __restrict__ labels,
                                                     float* __restrict__ fc,
                                                     float* __restrict__ nums) {
  const int m = blockIdx.x;
  const int c = labels[m];
  const float* src = feats + (size_t)m * D_DIM;
  float* dst = fc + (size_t)c * D_DIM;
  for (int d = threadIdx.x; d < D_DIM; d += 256) {
    atomicAdd(&dst[d], src[d]);
  }
  if (threadIdx.x == 0) atomicAdd(&nums[c], 1.0f);
}

// ---------------------------------------------------------------------------
// Kernel 3: S[b,c] = dot(x_norm[b,:], Fc[c,:]) via V_WMMA_F32_16X16X4_F32.
// Wave tile: 16(M) x 16(N); block = 8 waves -> 16 x 128 tile.
// Grid: (B/16, C/128) = (64, 32).
//
// A-matrix (16x4 f32) lane layout per ISA 7.12.2:
//   lanes 0-15  : VGPR0=K0, VGPR1=K1   (row M = lane)
//   lanes 16-31 : VGPR0=K2, VGPR1=K3   (row M = lane-16)
// B-matrix (4x16) mirrored: N = lane%16, same K split.
// C/D 16x16 f32: vgpr i -> row (lane<16 ? i : i+8), col lane%16.
// ---------------------------------------------------------------------------
__global__ __launch_bounds__(256) void gemm_wmma(const float* __restrict__ X,
                                                 const float* __restrict__ Fc,
                                                 float* __restrict__ S) {
  const int wave = threadIdx.x >> 5;
  const int lane = threadIdx.x & 31;
  const int l16 = lane & 15;
  const int hi = lane >> 4;  // 0: lanes 0-15, 1: lanes 16-31

  const int tileM = blockIdx.x * 16;
  const int tileN = blockIdx.y * 128 + wave * 16;

  const float* arow = X + (size_t)(tileM + l16) * D_DIM;   // A row (M = l16)
  const float* brow = Fc + (size_t)(tileN + l16) * D_DIM;  // B col (N = l16), row-major Fc

  v8f acc = {};
#if defined(__has_builtin) && __has_builtin(__builtin_amdgcn_wmma_f32_16x16x4_f32)
  for (int k = 0; k < D_DIM; k += 4) {
    const int ka = k + hi * 2;
    v2f a;
    a.x = arow[ka];
    a.y = arow[ka + 1];
    v2f b;
    b.x = brow[ka];
    b.y = brow[ka + 1];
    acc = __builtin_amdgcn_wmma_f32_16x16x4_f32(false, a, false, b, (short)0, acc,
                                                false, false);
  }
  const int row0 = tileM + hi * 8;
  const int col = tileN + l16;
#pragma unroll
  for (int i = 0; i < 8; ++i) {
    S[(size_t)(row0 + i) * C_DIM + col] = acc[i];
  }
#else
  // Scalar fallback (should not be taken on gfx1250): lane computes 8 outputs (one column, 8 rows).
  // 16x16 tile striped: lane l16 -> col, hi*8+i -> rows
  const int col = tileN + l16;
#pragma unroll
  for (int i = 0; i < 8; ++i) {
    const int r = tileM + hi * 8 + i;
    float s = 0.0f;
    for (int k = 0; k < D_DIM; ++k) s += X[(size_t)r * D_DIM + k] * Fc[(size_t)col * D_DIM + k];
    S[(size_t)r * C_DIM + col] = s;
  }
#endif
}

// ---------------------------------------------------------------------------
// Kernel 4: masked softmax over clusters + NLL at target; mean via atomic.
// One 256-thread block per sample b. Reference does not max-subtract
// (values bounded by 1/TEMP = 20), so neither do we.
// ---------------------------------------------------------------------------
__global__ __launch_bounds__(256) void softmax_loss(const float* __restrict__ S,
                                                    const float* __restrict__ nums,
                                                    const int* __restrict__ indexes,
                                                    const int* __restrict__ labels,
                                                    float* __restrict__ out) {
  __shared__ float red[8];
  const int b = blockIdx.x;
  const float* srow = S + (size_t)b * C_DIM;
  float sum = 0.0f;
  for (int c = threadIdx.x; c < C_DIM; c += 256) {
    const float n = nums[c];
    const float denom = (n > 0.0f) ? n : 1.0f;
    const float v = srow[c] / (TEMP_F * denom);
    sum += (n > 0.0f) ? expf(v) : 0.0f;
  }
  for (int off = 16; off > 0; off >>= 1) sum += __shfl_down(sum, off, 32);
  const int lane = threadIdx.x & 31;
  const int wave = threadIdx.x >> 5;
  if (lane == 0) red[wave] = sum;
  __syncthreads();
  if (threadIdx.x == 0) {
    float total = 0.0f;
#pragma unroll
    for (int i = 0; i < 8; ++i) total += red[i];
    const int t = labels[indexes[b]];
    const float n = nums[t];
    const float denom = (n > 0.0f) ? n : 1.0f;
    const float v = srow[t] / (TEMP_F * denom);
    const float e = (n > 0.0f) ? expf(v) : 0.0f;
    const float prob = e / (total + EPS_F);
    const float lp = logf(prob + EPS_F);
    atomicAdd(out, -lp * (1.0f / (float)B_DIM));
  }
}

// ---------------------------------------------------------------------------
// Host-side launcher. Inputs: [inputs f32 B*D, indexes int B, features f32 M*D,
// labels int M]. Output: scalar f32 loss.
// ---------------------------------------------------------------------------
extern "C" void kernel_launch(void* const* d_in, const int* in_sizes, int n_in,
                              void* d_out, int out_size, void* d_ws, size_t ws_size,
                              hipStream_t stream) {
  const float* inputs = (const float*)d_in[0];
  const int* indexes = (const int*)d_in[1];
  const float* features = (const float*)d_in[2];
  const int* labels = (const int*)d_in[3];
  float* out = (float*)d_out;

  float* ws = (float*)d_ws;
  float* xn = ws;                        // B*D
  float* fc = xn + B_DIM * D_DIM;        // C*D
  float* nums = fc + C_DIM * D_DIM;      // C
  float* S = nums + C_DIM;               // B*C

  // 0) zero accumulators + output
  {
    const int n = C_DIM * D_DIM + C_DIM;
    zero_ws<<<(n + 255) / 256, 256, 0, stream>>>(fc, nums, out);
  }
  // 1) normalize inputs
  normalize_rows<<<B_DIM, 256, 0, stream>>>(inputs, xn);
  // 2) cluster-sum features (+counts)
  cluster_accum<<<M_DIM, 256, 0, stream>>>(features, labels, fc, nums);
  // 3) WMMA GEMM: S = xn @ fc^T
  {
    dim3 grid(B_DIM / 16, C_DIM / 128);
    gemm_wmma<<<grid, 256, 0, stream>>>(xn, fc, S);
  }
  // 4) masked softmax + NLL mean
  softmax_loss<<<B_DIM, 256, 0, stream>>>(S, nums, indexes, labels, out);
}